// LSTM_46823733460992
// MI455X (gfx1250) — compile-verified
//
#include <hip/hip_runtime.h>

// ---------------------------------------------------------------------------
// LSTM on gfx1250: persistent-kernel scan with fp32 WMMA (16x16x4).
//  - 64 WGs x 4 waves; WG = (batch row-block r, 2 hidden col-blocks)
//  - role-split waves: waves 0-1 compute x_t@W_xh (K=0..511, pre-staged x),
//    waves 2-3 compute h_{t-1}@W_hh (K=512..1023, freshly staged h); the
//    x-half overlaps the h-staging L2 round-trip. Partials meet in LDS.
//  - async global->LDS staging (ASYNCcnt); x double-buffered a step ahead
//  - LDS rows padded (+4 floats): A-fragment ds_load_b64 is bank-conflict-free
//  - per-role LDS split-barrier; per-r-group global barrier (block-diagonal dep)
//  - weights packed once into per-lane B-fragment order, streamed from L2
//  - cell state c lives in the x-half wave's VGPRs for all 512 steps
// ---------------------------------------------------------------------------

typedef __attribute__((ext_vector_type(2))) float v2f;
typedef __attribute__((ext_vector_type(4))) float v4f;
typedef __attribute__((ext_vector_type(8))) float v8f;

#define LSTM_N 64
#define LSTM_T 512
#define LSTM_D 512
#define LSTM_H 512

// packed weights layout: [c:32][kc:256][lane:32][gate:4][v:2] floats
#define PACK_KC_STRIDE  (32 * 8)                 // floats per kc slice (1KB/wave)
#define PACK_C_STRIDE   (256 * PACK_KC_STRIDE)   // floats per hidden col-block
#define PACK_FLOATS     (32 * PACK_C_STRIDE)     // 2,097,152 floats = 8 MB
#define HBUF_FLOATS     (2 * LSTM_N * LSTM_H)    // double-buffered h, 256 KB

#define NUM_WG 64
#define WGS_PER_GROUP 16                         // WGs per batch row-group
#define WG_THREADS 128                           // 4 waves
#define ROLE_THREADS 64                          // 2 waves per role

#define LDS_ROW (LSTM_D + 4)                     // 516: bank-conflict skew

__device__ __forceinline__ v8f wmma_f32_k4(v2f a, v2f b, v8f c) {
  // 8 args: (neg_a, A, neg_b, B, c_mod, C, reuse_a, reuse_b)
  return __builtin_amdgcn_wmma_f32_16x16x4_f32(false, a, false, b, (short)0, c,
                                               false, false);
}

__device__ __forceinline__ float sigm(float v) {
  return 1.0f / (1.0f + __expf(-v));
}
__device__ __forceinline__ float tanh_fast(float v) {
  float e = __expf(-2.0f * fabsf(v));
  float r = (1.0f - e) / (1.0f + e);
  return copysignf(r, v);
}

// LDS byte offset of a shared-memory pointer (flat LDS aperture: low 32 bits)
__device__ __forceinline__ unsigned lds_off(const void* p) {
  return (unsigned)(uintptr_t)p;
}

// Async copy 16B global -> LDS, tracked by ASYNCcnt (per-lane transfer).
__device__ __forceinline__ void async_g2l_b128(unsigned lds, const void* g) {
  asm volatile("global_load_async_to_lds_b128 %0, %1, off"
               :: "v"(lds), "v"((unsigned long long)(uintptr_t)g)
               : "memory");
}
__device__ __forceinline__ void wait_asynccnt0() {
  asm volatile("s_wait_asynccnt 0x0" ::: "memory");
}
// WGP-scope prefetch (default scope 0: fill all cache levels incl. WGP$).
__device__ __forceinline__ void prefetch_wgp(const void* p) {
  asm volatile("global_prefetch_b8 %0, off"
               :: "v"((unsigned long long)(uintptr_t)p) : "memory");
}

// Split-barrier between the two waves of one role (monotonic LDS counter).
__device__ __forceinline__ void role_barrier(unsigned* flag, int lane,
                                             unsigned target) {
  if (lane == 0) {
    __hip_atomic_fetch_add(flag, 1u, __ATOMIC_RELEASE,
                           __HIP_MEMORY_SCOPE_WORKGROUP);
  }
  while (__hip_atomic_load(flag, __ATOMIC_ACQUIRE,
                           __HIP_MEMORY_SCOPE_WORKGROUP) < target) {
  }
}

// Stage one 16x512 x row-block (timestep tsrc) into an LDS buffer.
// Called by the 64 x-role threads (tlocal in [0,64)).
__device__ __forceinline__ void stage_x(const float* __restrict__ x, int rbase,
                                        int tsrc, float (*xbuf)[LDS_ROW],
                                        int tlocal) {
  for (int chunk = tlocal; chunk < 2048; chunk += ROLE_THREADS) {
    int row = chunk >> 7;
    int col = (chunk & 127) << 2;  // float index, 16B granules
    const float* gx =
        x + ((size_t)(rbase + row) * LSTM_T + tsrc) * LSTM_D + col;
    async_g2l_b128(lds_off(&xbuf[row][col]), gx);
  }
}

// ---------------------------------------------------------------------------
// Pack [W_xh ; W_hh] (K=1024 x 2048) into per-lane WMMA B-fragment order.
// B tile is 4(K) x 16(N): lane<16 holds K=0,1 (v=0,1), lane>=16 holds K=2,3,
// mirroring the documented f32 A-matrix layout. 4 gate fragments interleaved
// per lane so the scan kernel reads them with two b128 loads.
// ---------------------------------------------------------------------------
__global__ void lstm_pack_weights(const float* __restrict__ wxh,
                                  const float* __restrict__ whh,
                                  float* __restrict__ packed) {
  int idx = blockIdx.x * blockDim.x + threadIdx.x;
  if (idx >= PACK_FLOATS) return;
  int v    = idx & 1;
  int gate = (idx >> 1) & 3;
  int lane = (idx >> 3) & 31;
  int kc   = (idx >> 8) & 255;
  int c    = idx >> 16;  // 0..31
  int k    = kc * 4 + ((lane >> 4) << 1) + v;      // 0..1023
  int col  = gate * LSTM_H + c * 16 + (lane & 15); // 0..2047
  float w = (k < LSTM_D) ? wxh[(size_t)k * (4 * LSTM_H) + col]
                         : whh[(size_t)(k - LSTM_D) * (4 * LSTM_H) + col];
  packed[idx] = w;
}

// ---------------------------------------------------------------------------
// Zero h ping-pong buffers and the 4 per-r-group barrier counters.
// ---------------------------------------------------------------------------
__global__ void lstm_init_state(float* __restrict__ hbuf,
                                unsigned* __restrict__ cnt) {
  int i = blockIdx.x * blockDim.x + threadIdx.x;
  if (i < HBUF_FLOATS) hbuf[i] = 0.0f;
  if (i < 4) cnt[i] = 0u;
}

// ---------------------------------------------------------------------------
// Persistent scan kernel.
//   wg: r = wg>>4 (batch row-block), cpair = wg&15 (two hidden col-blocks).
//   wave wv: cidx = wv&1 (which col-block), khalf = wv>>1 (x-half / h-half).
// ---------------------------------------------------------------------------
__global__ __launch_bounds__(WG_THREADS, 1)
void lstm_scan(const float* __restrict__ x,
               const float* __restrict__ packed,
               const float* __restrict__ bias_xh,
               const float* __restrict__ bias_hh,
               float* __restrict__ hbuf,
               unsigned* __restrict__ cnt,
               float* __restrict__ out) {
  __shared__ float xs[2][16][LDS_ROW];      // 2 x 33 KB: x row-blocks
  __shared__ float hs[16][LDS_ROW];         // 33 KB: h_{t-1} row-block
  __shared__ float red[2][4][32][8];        // 8 KB: h-half partial gate sums
  __shared__ unsigned roleflag[2];          // split-barrier counters

  const int tid   = threadIdx.x;
  const int lane  = tid & 31;
  const int wv    = tid >> 5;                        // 0..3
  const int cidx  = wv & 1;                          // col-block within pair
  const int khalf = wv >> 1;                         // 0: x-half, 1: h-half
  const int tloc  = tid & (ROLE_THREADS - 1);        // thread id within role
  const int r     = blockIdx.x >> 4;                 // batch row-block 0..3
  const int c     = (blockIdx.x & 15) * 2 + cidx;    // hidden col-block 0..31
  const int rbase = r * 16;
  const int m     = lane & 15;                       // A row within tile
  const int klo   = (lane >> 4) << 1;                // A k-pair select
  const int j     = c * 16 + m;                      // this lane's hidden unit

  // gate pre-activation bias (x-half owns it; splat into its accumulators)
  const float b0 = bias_xh[0 * LSTM_H + j] + bias_hh[0 * LSTM_H + j];
  const float b1 = bias_xh[1 * LSTM_H + j] + bias_hh[1 * LSTM_H + j];
  const float b2 = bias_xh[2 * LSTM_H + j] + bias_hh[2 * LSTM_H + j];
  const float b3 = bias_xh[3 * LSTM_H + j] + bias_hh[3 * LSTM_H + j];

  const float* pbase = packed + (size_t)c * PACK_C_STRIDE + lane * 8;

  v8f cst = {0.f, 0.f, 0.f, 0.f, 0.f, 0.f, 0.f, 0.f};  // cell state (x-half)

  if (tid < 2) roleflag[tid] = 0u;
  __syncthreads();

  // Pre-stage x_0 into buffer 0 (drained by step 0's asynccnt wait).
  if (khalf == 0) stage_x(x, rbase, 0, xs[0], tloc);

  for (int t = 0; t < LSTM_T; ++t) {
    const float* hr = hbuf + (size_t)(t & 1) * (LSTM_N * LSTM_H);
    float*       hw = hbuf + (size_t)((t + 1) & 1) * (LSTM_N * LSTM_H);

    // ---- h-half stages h_{t-1}; x-half proceeds straight to compute ----
    if (khalf == 1) {
      for (int chunk = tloc; chunk < 2048; chunk += ROLE_THREADS) {
        int row = chunk >> 7;
        int col = (chunk & 127) << 2;
        const float* gh = hr + (size_t)(rbase + row) * LSTM_H + col;
        async_g2l_b128(lds_off(&hs[row][col]), gh);
      }
    }
    wait_asynccnt0();                        // drain own role's async loads
    role_barrier(&roleflag[khalf], lane, 2u * (unsigned)(t + 1));

    v8f A0, A1, A2, A3;
    if (khalf == 0) {
      A0 = (v8f){b0, b0, b0, b0, b0, b0, b0, b0};
      A1 = (v8f){b1, b1, b1, b1, b1, b1, b1, b1};
      A2 = (v8f){b2, b2, b2, b2, b2, b2, b2, b2};
      A3 = (v8f){b3, b3, b3, b3, b3, b3, b3, b3};
      const float (*xc)[LDS_ROW] = xs[t & 1];
      // ---- K = 0..511 : x_t @ W_xh ----
#pragma unroll 4
      for (int kc = 0; kc < 128; ++kc) {
        const float* bp = pbase + (size_t)kc * PACK_KC_STRIDE;
        prefetch_wgp(bp + 16 * PACK_KC_STRIDE);
        v4f w01 = *(const v4f*)bp;
        v4f w23 = *(const v4f*)(bp + 4);
        v2f a = *(const v2f*)&xc[m][kc * 4 + klo];
        v2f g0 = {w01[0], w01[1]}, g1 = {w01[2], w01[3]};
        v2f g2 = {w23[0], w23[1]}, g3 = {w23[2], w23[3]};
        A0 = wmma_f32_k4(a, g0, A0);
        A1 = wmma_f32_k4(a, g1, A1);
        A2 = wmma_f32_k4(a, g2, A2);
        A3 = wmma_f32_k4(a, g3, A3);
      }
      // Prefetch x_{t+1}; overlaps reduce + epilogue + barrier.
      if (t + 1 < LSTM_T) stage_x(x, rbase, t + 1, xs[(t + 1) & 1], tloc);
    } else {
      A0 = (v8f){0.f, 0.f, 0.f, 0.f, 0.f, 0.f, 0.f, 0.f};
      A1 = A0; A2 = A0; A3 = A0;
      // ---- K = 512..1023 : h_{t-1} @ W_hh ----
#pragma unroll 4
      for (int kc = 128; kc < 256; ++kc) {
        const float* bp = pbase + (size_t)kc * PACK_KC_STRIDE;
        prefetch_wgp(bp + 16 * PACK_KC_STRIDE);
        v4f w01 = *(const v4f*)bp;
        v4f w23 = *(const v4f*)(bp + 4);
        v2f a = *(const v2f*)&hs[m][(kc - 128) * 4 + klo];
        v2f g0 = {w01[0], w01[1]}, g1 = {w01[2], w01[3]};
        v2f g2 = {w23[0], w23[1]}, g3 = {w23[2], w23[3]};
        A0 = wmma_f32_k4(a, g0, A0);
        A1 = wmma_f32_k4(a, g1, A1);
        A2 = wmma_f32_k4(a, g2, A2);
        A3 = wmma_f32_k4(a, g3, A3);
      }
      // Publish partial gate sums for the paired x-half wave.
      v4f s;
      s = (v4f){A0[0], A0[1], A0[2], A0[3]}; *(v4f*)&red[cidx][0][lane][0] = s;
      s = (v4f){A0[4], A0[5], A0[6], A0[7]}; *(v4f*)&red[cidx][0][lane][4] = s;
      s = (v4f){A1[0], A1[1], A1[2], A1[3]}; *(v4f*)&red[cidx][1][lane][0] = s;
      s = (v4f){A1[4], A1[5], A1[6], A1[7]}; *(v4f*)&red[cidx][1][lane][4] = s;
      s = (v4f){A2[0], A2[1], A2[2], A2[3]}; *(v4f*)&red[cidx][2][lane][0] = s;
      s = (v4f){A2[4], A2[5], A2[6], A2[7]}; *(v4f*)&red[cidx][2][lane][4] = s;
      s = (v4f){A3[0], A3[1], A3[2], A3[3]}; *(v4f*)&red[cidx][3][lane][0] = s;
      s = (v4f){A3[4], A3[5], A3[6], A3[7]}; *(v4f*)&red[cidx][3][lane][4] = s;
    }

    __syncthreads();  // partials visible; both halves joined

    float hval[8];
    if (khalf == 0) {
      // ---- reduce partials + lane-local cell update ----
      v4f lo, hi;
      lo = *(const v4f*)&red[cidx][0][lane][0];
      hi = *(const v4f*)&red[cidx][0][lane][4];
#pragma unroll
      for (int e = 0; e < 4; ++e) { A0[e] += lo[e]; A0[e + 4] += hi[e]; }
      lo = *(const v4f*)&red[cidx][1][lane][0];
      hi = *(const v4f*)&red[cidx][1][lane][4];
#pragma unroll
      for (int e = 0; e < 4; ++e) { A1[e] += lo[e]; A1[e + 4] += hi[e]; }
      lo = *(const v4f*)&red[cidx][2][lane][0];
      hi = *(const v4f*)&red[cidx][2][lane][4];
#pragma unroll
      for (int e = 0; e < 4; ++e) { A2[e] += lo[e]; A2[e + 4] += hi[e]; }
      lo = *(const v4f*)&red[cidx][3][lane][0];
      hi = *(const v4f*)&red[cidx][3][lane][4];
#pragma unroll
      for (int e = 0; e < 4; ++e) { A3[e] += lo[e]; A3[e + 4] += hi[e]; }

#pragma unroll
      for (int e = 0; e < 8; ++e) {
        float iv = sigm(A0[e]);
        float fv = sigm(A1[e]);
        float gv = tanh_fast(A2[e]);
        float ov = sigm(A3[e]);
        float cn = fmaf(fv, cst[e], iv * gv);
        cst[e] = cn;
        float hv = ov * tanh_fast(cn);
        hval[e] = hv;
        int n = rbase + ((lane < 16) ? e : (e + 8));
        hw[(size_t)n * LSTM_H + j] = hv;  // consumed next step via staging
      }
    }

    // ---- r-group barrier (16 WGs): release h, wait producers, acquire ----
    __threadfence();
    __syncthreads();
    if (tid == 0) {
      atomicAdd(&cnt[r], 1u);
      const unsigned target = (unsigned)(t + 1) * WGS_PER_GROUP;
      while (__hip_atomic_load(&cnt[r], __ATOMIC_ACQUIRE,
                               __HIP_MEMORY_SCOPE_AGENT) < target) {
        __builtin_amdgcn_s_sleep(1);
      }
    }
    __syncthreads();
    __threadfence();

    // y writes off the critical path: retired by the NEXT step's fence.
    if (khalf == 0) {
#pragma unroll
      for (int e = 0; e < 8; ++e) {
        int n = rbase + ((lane < 16) ? e : (e + 8));
        out[((size_t)n * LSTM_T + t) * LSTM_H + j] = hval[e];
      }
    }
  }
}

// ---------------------------------------------------------------------------
extern "C" void kernel_launch(void* const* d_in, const int* in_sizes, int n_in,
                              void* d_out, int out_size, void* d_ws,
                              size_t ws_size, hipStream_t stream) {
  const float* x   = (const float*)d_in[0];  // (64, 512, 512)
  const float* wxh = (const float*)d_in[1];  // (512, 2048)
  const float* whh = (const float*)d_in[2];  // (512, 2048)
  const float* bxh = (const float*)d_in[3];  // (2048,)
  const float* bhh = (const float*)d_in[4];  // (2048,)
  float* out = (float*)d_out;                // (64, 512, 512)

  char* ws = (char*)d_ws;
  float*    packed = (float*)ws;                                    // 8 MB
  float*    hbuf   = (float*)(ws + (size_t)PACK_FLOATS * 4);        // 256 KB
  unsigned* cnt    = (unsigned*)(ws + (size_t)PACK_FLOATS * 4 +
                                 (size_t)HBUF_FLOATS * 4);          // 16 B

  lstm_pack_weights<<<PACK_FLOATS / 256, 256, 0, stream>>>(wxh, whh, packed);
  lstm_init_state<<<(HBUF_FLOATS + 255) / 256, 256, 0, stream>>>(hbuf, cnt);
  lstm_scan<<<NUM_WG, WG_THREADS, 0, stream>>>(x, packed, bxh, bhh, hbuf, cnt,
                                               out);
}